// aagmm_layer_77567109366515
// MI455X (gfx1250) — compile-verified
//
#include <hip/hip_runtime.h>
#include <math.h>

// GMM log-responsibilities on gfx1250 (MI455X), f32 WMMA path.
//   BATCH=4096, K=256 classes, D=256 dims, all f32.
//
// dist_sq[b,k] = sum_d (x-c)^2 * Dinv  ==  [x | x^2] @ [c*Dinv | -0.5*Dinv]^T  (up to bias)
// -> single 4096 x 256 x 512 f32 GEMM via V_WMMA_F32_16X16X4_F32, fused softmax-log epilogue.

#define BATCH 4096
#define NCLS  256
#define DIMS  256
#define KDIM  512            // concatenated K-dim: [x | x^2]
#define EPSV  1e-8f
#define XS_STRIDE 516        // 256*2 + 4 pad floats -> conflict-free ds_load_b64

typedef __attribute__((ext_vector_type(2))) float v2f;
typedef __attribute__((ext_vector_type(4))) float v4f;
typedef __attribute__((ext_vector_type(8))) float v8f;

// ---------------------------------------------------------------------------
// Prep 1: per-class bias[k] = dsf[k] - max_k(dsf) - 0.5*c2[k]
//   dsf[k] = -0.5 * sum_d log(|D|+eps),  c2[k] = sum_d c^2/( |D|+eps )
// One block of 256 threads, one class per thread.
// ---------------------------------------------------------------------------
__global__ __launch_bounds__(256) void gmm_prep_bias(const float* __restrict__ centers,
                                                     const float* __restrict__ D,
                                                     float* __restrict__ bias) {
    int k = threadIdx.x;
    float ld = 0.0f, c2 = 0.0f;
    const float* dk = D + k * DIMS;
    const float* ck = centers + k * DIMS;
    for (int d = 0; d < DIMS; ++d) {
        float da = fabsf(dk[d]) + EPSV;
        ld += logf(da);
        float c = ck[d];
        c2 += c * c / da;
    }
    float dsf = -0.5f * ld;
    __shared__ float sm[256];
    sm[k] = dsf;
    __syncthreads();
    for (int off = 128; off > 0; off >>= 1) {
        if (k < off) sm[k] = fmaxf(sm[k], sm[k + off]);
        __syncthreads();
    }
    float M = sm[0];
    bias[k] = dsf - M - 0.5f * c2;
}

// ---------------------------------------------------------------------------
// Prep 2: pack B' = [ c*Dinv | -0.5*Dinv ]  (shape [K=256][KDIM=512]) directly
// into WMMA B-fragment order:  element index
//   i = (((ct*64 + p)*32 + lane)*4) + (s_parity*2 + v)
// where class k = ct*16 + (lane&15), K-step s = 2p + s_parity,
// j = 4s + 2*(lane>=16) + v.  Consumer then does one b128 load per lane
// covering two K-steps. Total 131072 elements (512 KB).
// ---------------------------------------------------------------------------
__global__ __launch_bounds__(256) void gmm_pack_bfrag(const float* __restrict__ centers,
                                                      const float* __restrict__ D,
                                                      float* __restrict__ Bf) {
    int i = blockIdx.x * 256 + threadIdx.x;       // 0 .. 131071
    int v    = i & 1;
    int half = (i >> 1) & 1;
    int lane = (i >> 2) & 31;
    int p    = (i >> 7) & 63;
    int ct   = i >> 13;
    int s  = 2 * p + half;
    int kk = 2 * (lane >> 4) + v;
    int j  = 4 * s + kk;
    int k  = ct * 16 + (lane & 15);
    float val;
    if (j < DIMS) {
        float da = fabsf(D[k * DIMS + j]) + EPSV;
        val = centers[k * DIMS + j] / da;          // pairs with x
    } else {
        int d = j - DIMS;
        float da = fabsf(D[k * DIMS + d]) + EPSV;
        val = -0.5f / da;                          // pairs with x^2
    }
    Bf[i] = val;
}

// ---------------------------------------------------------------------------
// Main: 16-row tile per block, 8 waves x 32 classes, f32 WMMA GEMM + fused
// softmax-log epilogue.
// ---------------------------------------------------------------------------
__global__ __launch_bounds__(256) void gmm_main(const float* __restrict__ x,
                                                const float* __restrict__ Bf,
                                                const float* __restrict__ bias,
                                                float* __restrict__ out) {
    __shared__ float xs[16 * XS_STRIDE];   // [x | x^2] tile, padded rows
    __shared__ float lg[16 * NCLS];        // logits tile
    __shared__ float redmax[16 * 16];
    __shared__ float redsum[16 * 16];

    const int t  = threadIdx.x;
    const int b0 = blockIdx.x * 16;

    // Stage x rows + squares into LDS (coalesced: col = t).
    for (int r = 0; r < 16; ++r) {
        float v = x[(b0 + r) * DIMS + t];
        xs[r * XS_STRIDE + t]        = v;
        xs[r * XS_STRIDE + DIMS + t] = v * v;
    }
    __syncthreads();

    const int w    = t >> 5;        // wave 0..7
    const int lane = t & 31;
    const int cb   = w * 32;        // 32 classes per wave (two 16-wide tiles)
    const int row  = lane & 15;     // A-matrix M row for this lane
    const int kkb  = (lane >> 4) * 2;

    const v4f* B0 = (const v4f*)(Bf + (size_t)(2 * w)     * 64 * 128);
    const v4f* B1 = (const v4f*)(Bf + (size_t)(2 * w + 1) * 64 * 128);
    const float* arow = xs + row * XS_STRIDE + kkb;

    v8f acc0 = {};
    v8f acc1 = {};

#pragma unroll 4
    for (int p = 0; p < 64; ++p) {           // 64 K-step pairs = K-dim 512
        v2f a01 = *(const v2f*)(arow + 8 * p);       // K-step 2p   (k0,k1)
        v2f a23 = *(const v2f*)(arow + 8 * p + 4);   // K-step 2p+1 (k0,k1)
        v4f b0  = B0[p * 32 + lane];                 // two steps, coalesced b128
        v4f b1  = B1[p * 32 + lane];
        v2f b0lo = {b0.x, b0.y}, b0hi = {b0.z, b0.w};
        v2f b1lo = {b1.x, b1.y}, b1hi = {b1.z, b1.w};
        acc0 = __builtin_amdgcn_wmma_f32_16x16x4_f32(false, a01, false, b0lo,
                                                     (short)0, acc0, false, false);
        acc0 = __builtin_amdgcn_wmma_f32_16x16x4_f32(false, a23, false, b0hi,
                                                     (short)0, acc0, false, false);
        acc1 = __builtin_amdgcn_wmma_f32_16x16x4_f32(false, a01, false, b1lo,
                                                     (short)0, acc1, false, false);
        acc1 = __builtin_amdgcn_wmma_f32_16x16x4_f32(false, a23, false, b1hi,
                                                     (short)0, acc1, false, false);
    }

    // Spill logits (+ per-class bias) to LDS.
    // C/D layout: vgpr i, lane L -> M = i + 8*(L>=16), N = L&15.
    const int n     = lane & 15;
    const int mhalf = (lane >> 4) * 8;
    const float bias0 = bias[cb + n];
    const float bias1 = bias[cb + 16 + n];
    for (int i = 0; i < 8; ++i) {
        lg[(mhalf + i) * NCLS + cb + n]      = acc0[i] + bias0;
        lg[(mhalf + i) * NCLS + cb + 16 + n] = acc1[i] + bias1;
    }
    __syncthreads();

    // Softmax-log over 256 classes per row; 16 threads per row, 16 cols each.
    const int rrow = t >> 4;
    const int seg  = t & 15;
    const float* lrow = lg + rrow * NCLS + seg * 16;

    float mx = -1e30f;
    for (int i = 0; i < 16; ++i) mx = fmaxf(mx, lrow[i]);
    redmax[rrow * 16 + seg] = mx;
    __syncthreads();

    float R = -1e30f;
    for (int i = 0; i < 16; ++i) R = fmaxf(R, redmax[rrow * 16 + i]);

    float se = 0.0f;
    for (int i = 0; i < 16; ++i) se += expf(lrow[i] - R);
    redsum[rrow * 16 + seg] = se;
    __syncthreads();

    float denom = 0.0f;
    for (int i = 0; i < 16; ++i) denom += redsum[rrow * 16 + i];
    float inv = 1.0f / denom;

    float* orow = out + (size_t)(b0 + rrow) * NCLS + seg * 16;
    for (int i = 0; i < 16; ++i) {
        float pr = expf(lrow[i] - R) * inv;
        pr = fmaxf(pr, EPSV);
        orow[i] = logf(pr);
    }
}

// ---------------------------------------------------------------------------
extern "C" void kernel_launch(void* const* d_in, const int* in_sizes, int n_in,
                              void* d_out, int out_size, void* d_ws, size_t ws_size,
                              hipStream_t stream) {
    const float* x       = (const float*)d_in[0];   // [4096,256]
    const float* centers = (const float*)d_in[1];   // [256,256]
    const float* D       = (const float*)d_in[2];   // [256,256]
    float* out = (float*)d_out;                     // [4096,256]

    float* Bf   = (float*)d_ws;                     // 131072 floats (512 KB)
    float* bias = (float*)d_ws + NCLS * KDIM;       // 256 floats

    gmm_prep_bias<<<1, 256, 0, stream>>>(centers, D, bias);
    gmm_pack_bfrag<<<(NCLS * KDIM) / 256, 256, 0, stream>>>(centers, D, Bf);
    gmm_main<<<BATCH / 16, 256, 0, stream>>>(x, Bf, bias, out);
}